// APPNProp_3178275799577
// MI455X (gfx1250) — compile-verified
//
#include <hip/hip_runtime.h>
#include <hip/hip_bf16.h>
#include <stdint.h>

#define N_NODES_C 50000
#define N_EDGES_C 800000
#define D_FEAT_C  64
#define ALPHA_C   0.1f
#define K_ITERS_C 10

#define WPB   8            // waves per block (wave32)
#define TPB   (WPB * 32)
#define CHUNK 32           // edges staged per async chunk (one per lane)

// ---- CDNA5 async-to-LDS path (device pass only) ----
#if defined(__HIP_DEVICE_COMPILE__) && defined(__gfx1250__) && \
    __has_builtin(__builtin_amdgcn_global_load_async_to_lds_b32) && \
    __has_builtin(__builtin_amdgcn_s_wait_asynccnt)
#define USE_ASYNC 1
#else
#define USE_ASYNC 0
#endif

#if USE_ASYNC
typedef __attribute__((address_space(1))) int g_int_t;   // global (AS1)
typedef __attribute__((address_space(3))) int l_int_t;   // LDS (AS3)

__device__ __forceinline__ l_int_t* to_lds(void* p) {
    // flat->LDS: low 32 bits of the flat address are the LDS byte offset
    return (l_int_t*)(uint32_t)(uintptr_t)p;
}
__device__ __forceinline__ g_int_t* to_gbl(const void* p) {
    return (g_int_t*)(uintptr_t)p;
}
#endif

// ---------------- preprocessing: build CSR on device ----------------

__global__ void zero_i32_kernel(int* __restrict__ p, int n) {
    int i = blockIdx.x * blockDim.x + threadIdx.x;
    if (i < n) p[i] = 0;
}

__global__ void count_deg_kernel(const int* __restrict__ edge_dst,
                                 int* __restrict__ deg, int n_edges) {
    int i = blockIdx.x * blockDim.x + threadIdx.x;
    if (i < n_edges) atomicAdd(&deg[edge_dst[i]], 1);
}

// single-block exclusive scan over deg[0..n) -> rowptr[0..n], nextp copy
__global__ __launch_bounds__(1024)
void scan_rowptr_kernel(const int* __restrict__ deg, int* __restrict__ rowptr,
                        int* __restrict__ nextp, int n) {
    __shared__ int sums[1024];
    const int t   = threadIdx.x;
    const int per = (n + 1023) / 1024;
    const int s0  = t * per;
    const int s1  = (s0 + per < n) ? (s0 + per) : n;

    int s = 0;
    for (int i = s0; i < s1; ++i) s += deg[i];
    sums[t] = s;
    __syncthreads();
    // Hillis-Steele inclusive scan
    for (int off = 1; off < 1024; off <<= 1) {
        int v = (t >= off) ? sums[t - off] : 0;
        __syncthreads();
        sums[t] += v;
        __syncthreads();
    }
    int run = (t == 0) ? 0 : sums[t - 1];
    for (int i = s0; i < s1; ++i) {
        rowptr[i] = run;
        nextp[i]  = run;
        run += deg[i];
    }
    if (t == 1023) rowptr[n] = sums[1023];   // grand total
}

__global__ void scatter_edges_kernel(const int* __restrict__ edge_src,
                                     const int* __restrict__ edge_dst,
                                     const float* __restrict__ edge_w,
                                     int* __restrict__ nextp,
                                     int* __restrict__ csr_src,
                                     float* __restrict__ csr_w, int n_edges) {
    int i = blockIdx.x * blockDim.x + threadIdx.x;
    if (i < n_edges) {
        int d   = edge_dst[i];
        int pos = atomicAdd(&nextp[d], 1);
        csr_src[pos] = edge_src[i];
        csr_w[pos]   = edge_w[i];
    }
}

// ---------------- propagation: one wave32 per destination row ----------------
// lane owns features {2*lane, 2*lane+1}; each edge gather is one coalesced
// 256B (b64/lane) row read, L2-resident. Edge stream is double-buffered into
// LDS via global_load_async_to_lds_b32 + s_wait_asynccnt (ASYNCcnt).

__global__ __launch_bounds__(TPB)
void appnp_prop_kernel(const float* __restrict__ h_in,
                       const float* __restrict__ x,
                       const int* __restrict__ rowptr,
                       const int* __restrict__ csr_src,
                       const float* __restrict__ csr_w,
                       float* __restrict__ h_out,
                       int n_nodes, int wave_stride) {
    __shared__ int   ls[WPB][2][CHUNK];
    __shared__ float lw[WPB][2][CHUNK];

    const int lane = threadIdx.x & 31;
    const int wv   = threadIdx.x >> 5;
    const int gw   = blockIdx.x * WPB + wv;

    for (int node = gw; node < n_nodes; node += wave_stride) {
        const int r0  = rowptr[node];
        const int r1  = rowptr[node + 1];
        const int deg = r1 - r0;
        const int nch = (deg + CHUNK - 1) >> 5;

        float a0 = 0.0f, a1 = 0.0f;

#if USE_ASYNC
        if (nch > 0) {
            const int e = r0 + lane;
            if (e < r1) {
                __builtin_amdgcn_global_load_async_to_lds_b32(
                    to_gbl(&csr_src[e]), to_lds(&ls[wv][0][lane]), 0, 0);
                __builtin_amdgcn_global_load_async_to_lds_b32(
                    to_gbl(&csr_w[e]),   to_lds(&lw[wv][0][lane]), 0, 0);
            }
        }
        for (int c = 0; c < nch; ++c) {
            const int cb = c & 1;
            if (c + 1 < nch) {
                const int e = r0 + (c + 1) * CHUNK + lane;
                if (e < r1) {
                    __builtin_amdgcn_global_load_async_to_lds_b32(
                        to_gbl(&csr_src[e]), to_lds(&ls[wv][cb ^ 1][lane]), 0, 0);
                    __builtin_amdgcn_global_load_async_to_lds_b32(
                        to_gbl(&csr_w[e]),   to_lds(&lw[wv][cb ^ 1][lane]), 0, 0);
                }
                __builtin_amdgcn_s_wait_asynccnt(2);  // chunk c complete (in-order)
            } else {
                __builtin_amdgcn_s_wait_asynccnt(0);  // drain
            }
            asm volatile("" ::: "memory");  // LDS written behind compiler's back

            const volatile int*   ps = &ls[wv][cb][0];
            const volatile float* pw = &lw[wv][cb][0];
            const int rem = deg - c * CHUNK;
            const int m   = (rem < CHUNK) ? rem : CHUNK;
            for (int j = 0; j < m; ++j) {
                const int   s   = ps[j];               // broadcast LDS read
                const float wgt = pw[j];
                const float2 hv = *(const float2*)&h_in[s * D_FEAT_C + 2 * lane];
                a0 = fmaf(wgt, hv.x, a0);
                a1 = fmaf(wgt, hv.y, a1);
            }
        }
#else
        for (int c = 0; c < nch; ++c) {
            const int e = r0 + c * CHUNK + lane;
            int sv = 0; float wvv = 0.0f;
            if (e < r1) { sv = csr_src[e]; wvv = csr_w[e]; }
            ls[wv][0][lane] = sv;
            lw[wv][0][lane] = wvv;
            const int rem = deg - c * CHUNK;
            const int m   = (rem < CHUNK) ? rem : CHUNK;
            for (int j = 0; j < m; ++j) {
                const int   s   = ls[wv][0][j];
                const float wgt = lw[wv][0][j];
                const float2 hv = *(const float2*)&h_in[s * D_FEAT_C + 2 * lane];
                a0 = fmaf(wgt, hv.x, a0);
                a1 = fmaf(wgt, hv.y, a1);
            }
        }
#endif
        const float2 xv = *(const float2*)&x[node * D_FEAT_C + 2 * lane];
        float2 out;
        out.x = (1.0f - ALPHA_C) * a0 + ALPHA_C * xv.x;
        out.y = (1.0f - ALPHA_C) * a1 + ALPHA_C * xv.y;
        *(float2*)&h_out[node * D_FEAT_C + 2 * lane] = out;
    }
}

// ---------------- host side ----------------

extern "C" void kernel_launch(void* const* d_in, const int* in_sizes, int n_in,
                              void* d_out, int out_size, void* d_ws, size_t ws_size,
                              hipStream_t stream) {
    const float* x    = (const float*)d_in[0];
    const int*   esrc = (const int*)d_in[1];
    const int*   edst = (const int*)d_in[2];
    const float* ew   = (const float*)d_in[3];

    char*  ws  = (char*)d_ws;
    size_t off = 0;
    auto alloc = [&](size_t bytes) -> void* {
        void* p = ws + off;
        off = (off + bytes + 255) & ~(size_t)255;
        return p;
    };
    int*   deg     = (int*)  alloc((size_t)N_NODES_C * 4);
    int*   rowptr  = (int*)  alloc((size_t)(N_NODES_C + 1) * 4);
    int*   nextp   = (int*)  alloc((size_t)N_NODES_C * 4);
    int*   csr_src = (int*)  alloc((size_t)N_EDGES_C * 4);
    float* csr_w   = (float*)alloc((size_t)N_EDGES_C * 4);
    float* h_tmp   = (float*)alloc((size_t)N_NODES_C * D_FEAT_C * 4);
    (void)ws_size; (void)in_sizes; (void)n_in; (void)out_size;

    // Build CSR (deterministic work; done fresh every call)
    zero_i32_kernel<<<(N_NODES_C + 255) / 256, 256, 0, stream>>>(deg, N_NODES_C);
    count_deg_kernel<<<(N_EDGES_C + 255) / 256, 256, 0, stream>>>(edst, deg, N_EDGES_C);
    scan_rowptr_kernel<<<1, 1024, 0, stream>>>(deg, rowptr, nextp, N_NODES_C);
    scatter_edges_kernel<<<(N_EDGES_C + 255) / 256, 256, 0, stream>>>(
        esrc, edst, ew, nextp, csr_src, csr_w, N_EDGES_C);

    // K iterations, ping-pong so iteration 10 (even count) lands in d_out
    float*       hout    = (float*)d_out;
    const float* hin     = x;
    float*       bufs[2] = { h_tmp, hout };
    const int    blocks  = 1024;
    const int    wstride = blocks * WPB;
    for (int k = 0; k < K_ITERS_C; ++k) {
        float* out = bufs[k & 1];
        appnp_prop_kernel<<<blocks, TPB, 0, stream>>>(
            hin, x, rowptr, csr_src, csr_w, out, N_NODES_C, wstride);
        hin = out;
    }
}